// GINEEncoder_70806830842064
// MI455X (gfx1250) — compile-verified
//
#include <hip/hip_runtime.h>
#include <hip/hip_bf16.h>

#define N_NODES 200000
#define E_EDGES 4000000
#define DIM     32
#define G_BATCH 512
#define OUTD    64
#define BN_EPS  1e-5f

typedef float v2f __attribute__((ext_vector_type(2)));
typedef float v8f __attribute__((ext_vector_type(8)));

__device__ __forceinline__ v8f wmma4(v2f a, v2f b, v8f c) {
    // V_WMMA_F32_16X16X4_F32 : D = A(16x4,f32) * B(4x16,f32) + C(16x16,f32)
    return __builtin_amdgcn_wmma_f32_16x16x4_f32(
        /*neg_a=*/false, a, /*neg_b=*/false, b,
        /*c_mod=*/(short)0, c, /*reuse_a=*/false, /*reuse_b=*/false);
}

// ---------------- edge aggregation: agg[dst] += relu(h[src] + emb[etype]) ----
// one "edge-wave": 32 lanes = 32 features of one edge (fully coalesced rows)
__global__ void __launch_bounds__(256)
edge_agg_kernel(const float* __restrict__ h, const int* __restrict__ src,
                const int* __restrict__ dst, const int* __restrict__ ety,
                const float* __restrict__ emb, float* __restrict__ agg)
{
    const int t    = blockIdx.x * 256 + threadIdx.x;
    const int e    = t >> 5;
    const int lane = t & 31;
    const int s  = src[e];
    const int d  = dst[e];
    const int ty = ety[e];
    const float msg = fmaxf(h[s * DIM + lane] + emb[ty * DIM + lane], 0.0f);
    atomicAdd(&agg[d * DIM + lane], msg);
}

// ---------------- GEMM1: t = (h+agg) @ W + b, accumulate column sum/sumsq ----
// 4 waves/block, each wave owns a 16-row tile; K=32 via 8x WMMA f32 16x16x4.
// NOTE: t may alias agg (each wave reads its own rows before writing them),
// so h2/agg2/t carry no __restrict__.
__global__ void __launch_bounds__(128)
gemm1_stats_kernel(const float* h, const float* agg,
                   const float* __restrict__ W, const float* __restrict__ bias,
                   float* t, float* __restrict__ colsum, float* __restrict__ colsumsq)
{
    __shared__ float sW[DIM * DIM];
    __shared__ float sb[DIM];
    __shared__ float ssum[DIM];
    __shared__ float ssq[DIM];
    const int tid = threadIdx.x;
    for (int i = tid; i < DIM * DIM; i += 128) sW[i] = W[i];
    if (tid < DIM) { sb[tid] = bias[tid]; ssum[tid] = 0.f; ssq[tid] = 0.f; }
    __syncthreads();

    const int lane = tid & 31;
    const int wave = tid >> 5;
    const int row0 = (blockIdx.x * 4 + wave) * 16;
    const int m    = lane & 15;
    const int half = lane >> 4;          // lanes 16..31 hold K+2,K+3 / rows M+8

    v8f acc0 = (v8f){0,0,0,0,0,0,0,0};
    v8f acc1 = acc0;
#pragma unroll
    for (int kb = 0; kb < 8; ++kb) {
        const int k0 = kb * 4 + half * 2;
        const int ia = (row0 + m) * DIM + k0;
        v2f a; a.x = h[ia] + agg[ia]; a.y = h[ia + 1] + agg[ia + 1];
        const int n = lane & 15;
        v2f b0, b1v;
        b0.x  = sW[k0 * DIM + n];        b0.y  = sW[(k0 + 1) * DIM + n];
        b1v.x = sW[k0 * DIM + n + 16];   b1v.y = sW[(k0 + 1) * DIM + n + 16];
        acc0 = wmma4(a, b0, acc0);
        acc1 = wmma4(a, b1v, acc1);
    }

    const int n = lane & 15;
    float s0 = 0.f, q0 = 0.f, s1 = 0.f, q1 = 0.f;
#pragma unroll
    for (int v = 0; v < 8; ++v) {
        const int row = row0 + v + half * 8;
        const float v0 = acc0[v] + sb[n];
        const float v1 = acc1[v] + sb[n + 16];
        t[row * DIM + n]      = v0;
        t[row * DIM + n + 16] = v1;
        s0 += v0; q0 += v0 * v0;
        s1 += v1; q1 += v1 * v1;
    }
    atomicAdd(&ssum[n], s0);      atomicAdd(&ssq[n], q0);
    atomicAdd(&ssum[n + 16], s1); atomicAdd(&ssq[n + 16], q1);
    __syncthreads();
    if (tid < DIM) {
        atomicAdd(&colsum[tid], ssum[tid]);
        atomicAdd(&colsumsq[tid], ssq[tid]);
    }
}

// ---------------- BN finalize: scale = g*rsig ; shift = be - scale*mu --------
__global__ void bn_finalize_kernel(const float* __restrict__ colsum,
                                   const float* __restrict__ colsumsq,
                                   const float* __restrict__ g,
                                   const float* __restrict__ be,
                                   float* __restrict__ scaleshift)
{
    const int c = threadIdx.x;
    const float invN = 1.0f / (float)N_NODES;
    const float mu  = colsum[c] * invN;
    const float var = colsumsq[c] * invN - mu * mu;
    const float rs  = rsqrtf(var + BN_EPS);
    const float sc  = g[c] * rs;
    scaleshift[c]       = sc;
    scaleshift[DIM + c] = be[c] - sc * mu;
}

// ---------------- GEMM2: out = [relu?]( relu(scale*t+shift) @ W + b ) -------
template<int NH, bool FINAL_RELU>
__global__ void __launch_bounds__(128)
gemm2_kernel(const float* __restrict__ t, const float* __restrict__ scaleshift,
             const float* __restrict__ W, const float* __restrict__ bias,
             float* __restrict__ out)
{
    constexpr int NC = 16 * NH;
    __shared__ float sW[DIM * NC];
    __shared__ float sb[NC];
    __shared__ float ssc[DIM];
    __shared__ float ssh[DIM];
    const int tid = threadIdx.x;
    for (int i = tid; i < DIM * NC; i += 128) sW[i] = W[i];
    if (tid < NC)  sb[tid] = bias[tid];
    if (tid < DIM) { ssc[tid] = scaleshift[tid]; ssh[tid] = scaleshift[DIM + tid]; }
    __syncthreads();

    const int lane = tid & 31;
    const int wave = tid >> 5;
    const int row0 = (blockIdx.x * 4 + wave) * 16;
    const int m    = lane & 15;
    const int half = lane >> 4;

    v8f acc[NH];
#pragma unroll
    for (int i = 0; i < NH; ++i) acc[i] = (v8f){0,0,0,0,0,0,0,0};

#pragma unroll
    for (int kb = 0; kb < 8; ++kb) {
        const int k0 = kb * 4 + half * 2;
        const int ia = (row0 + m) * DIM + k0;
        v2f a;
        a.x = fmaxf(ssc[k0]     * t[ia]     + ssh[k0],     0.0f);
        a.y = fmaxf(ssc[k0 + 1] * t[ia + 1] + ssh[k0 + 1], 0.0f);
        const int n = lane & 15;
#pragma unroll
        for (int hh = 0; hh < NH; ++hh) {
            v2f b;
            b.x = sW[k0 * NC + n + hh * 16];
            b.y = sW[(k0 + 1) * NC + n + hh * 16];
            acc[hh] = wmma4(a, b, acc[hh]);
        }
    }

    const int n = lane & 15;
#pragma unroll
    for (int v = 0; v < 8; ++v) {
        const int row = row0 + v + half * 8;
#pragma unroll
        for (int hh = 0; hh < NH; ++hh) {
            float val = acc[hh][v] + sb[n + hh * 16];
            if (FINAL_RELU) val = fmaxf(val, 0.0f);
            out[row * NC + n + hh * 16] = val;
        }
    }
}

// ---------------- pooling ---------------------------------------------------
__device__ __forceinline__ unsigned fenc(float f) {
    unsigned u = __float_as_uint(f);
    return (u & 0x80000000u) ? ~u : (u | 0x80000000u);
}
__device__ __forceinline__ float fdec(unsigned v) {
    unsigned u = (v & 0x80000000u) ? (v & 0x7FFFFFFFu) : ~v;
    return __uint_as_float(u);
}

__global__ void __launch_bounds__(256)
pool_init_kernel(unsigned* __restrict__ gmax, float* __restrict__ gsum,
                 float* __restrict__ gcnt)
{
    const int i = blockIdx.x * 256 + threadIdx.x;      // G*OUTD threads
    gmax[i] = 0x007FFFFFu;                             // fenc(-inf)
    gsum[i] = 0.0f;
    if (i < G_BATCH) gcnt[i] = 0.0f;
}

__global__ void __launch_bounds__(256)
pool_kernel(const float* __restrict__ h5, const int* __restrict__ batch,
            unsigned* __restrict__ gmax, float* __restrict__ gsum,
            float* __restrict__ gcnt)
{
    const int idx  = blockIdx.x * 256 + threadIdx.x;   // N*OUTD threads
    const int node = idx >> 6;
    const int f    = idx & 63;
    const int g    = batch[node];
    const float v  = h5[idx];
    atomicMax(&gmax[g * OUTD + f], fenc(v));
    atomicAdd(&gsum[g * OUTD + f], v);
    if (f == 0) atomicAdd(&gcnt[g], 1.0f);
}

__global__ void __launch_bounds__(256)
final_kernel(const unsigned* __restrict__ gmax, const float* __restrict__ gsum,
             const float* __restrict__ gcnt, float* __restrict__ out)
{
    const int idx = blockIdx.x * 256 + threadIdx.x;    // G*OUTD threads
    const int g = idx >> 6;
    const int f = idx & 63;
    out[g * (2 * OUTD) + f]        = fdec(gmax[idx]);
    out[g * (2 * OUTD) + OUTD + f] = gsum[idx] / fmaxf(gcnt[g], 1.0f);
}

// ---------------- launch -----------------------------------------------------
extern "C" void kernel_launch(void* const* d_in, const int* in_sizes, int n_in,
                              void* d_out, int out_size, void* d_ws, size_t ws_size,
                              hipStream_t stream) {
    const float* x     = (const float*)d_in[0];
    const int*   eidx  = (const int*)d_in[1];
    const int*   src   = eidx;
    const int*   dst   = eidx + E_EDGES;
    const int*   ety   = (const int*)d_in[2];
    const int*   batch = (const int*)d_in[3];
    const float* emb   = (const float*)d_in[4];
    const float* W1s   = (const float*)d_in[5];
    const float* b1s   = (const float*)d_in[6];
    const float* g1s   = (const float*)d_in[7];
    const float* be1s  = (const float*)d_in[8];
    const float* W2s   = (const float*)d_in[9];
    const float* b2s   = (const float*)d_in[10];
    const float* w15   = (const float*)d_in[11];
    const float* b15   = (const float*)d_in[12];
    const float* g5    = (const float*)d_in[13];
    const float* be5   = (const float*)d_in[14];
    const float* w25   = (const float*)d_in[15];
    const float* b25   = (const float*)d_in[16];
    float* out = (float*)d_out;

    // workspace carve-up (floats)
    float* p    = (float*)d_ws;
    float* bufA = p;                  p += (size_t)N_NODES * DIM;   // h ping
    float* bufB = p;                  p += (size_t)N_NODES * DIM;   // h pong
    float* agg  = p;                  p += (size_t)N_NODES * DIM;   // also 't'
    float* h5   = p;                  p += (size_t)N_NODES * OUTD;  // conv5 out
    float* colsum     = p;            p += DIM;
    float* colsumsq   = p;            p += DIM;
    float* scaleshift = p;            p += 2 * DIM;
    unsigned* gmax = (unsigned*)p;    p += (size_t)G_BATCH * OUTD;
    float* gsum = p;                  p += (size_t)G_BATCH * OUTD;
    float* gcnt = p;                  p += G_BATCH;
    float* tbuf = agg;                // t aliases agg (row-block exclusive)

    const float* hcur = x;
    float* bufs[2] = { bufA, bufB };

    const int edgeBlocks = (E_EDGES * 32) / 256;       // 500000
    const int gemmBlocks = N_NODES / 64;               // 3125 (4 tiles/block)

    for (int i = 0; i < 5; ++i) {
        hipMemsetAsync(agg, 0, (size_t)N_NODES * DIM * sizeof(float), stream);
        hipMemsetAsync(colsum, 0, 2 * DIM * sizeof(float), stream);  // sum+sumsq

        edge_agg_kernel<<<edgeBlocks, 256, 0, stream>>>(hcur, src, dst, ety, emb, agg);

        const float* W1 = (i < 4) ? W1s  + (size_t)i * DIM * DIM : w15;
        const float* B1 = (i < 4) ? b1s  + (size_t)i * DIM       : b15;
        const float* Gg = (i < 4) ? g1s  + (size_t)i * DIM       : g5;
        const float* Be = (i < 4) ? be1s + (size_t)i * DIM       : be5;

        gemm1_stats_kernel<<<gemmBlocks, 128, 0, stream>>>(hcur, agg, W1, B1,
                                                           tbuf, colsum, colsumsq);
        bn_finalize_kernel<<<1, DIM, 0, stream>>>(colsum, colsumsq, Gg, Be, scaleshift);

        if (i < 4) {
            gemm2_kernel<2, true><<<gemmBlocks, 128, 0, stream>>>(
                tbuf, scaleshift, W2s + (size_t)i * DIM * DIM, b2s + (size_t)i * DIM,
                bufs[i & 1]);
            hcur = bufs[i & 1];
        } else {
            gemm2_kernel<4, false><<<gemmBlocks, 128, 0, stream>>>(
                tbuf, scaleshift, w25, b25, h5);
        }
    }

    pool_init_kernel<<<(G_BATCH * OUTD) / 256, 256, 0, stream>>>(gmax, gsum, gcnt);
    pool_kernel<<<((size_t)N_NODES * OUTD) / 256, 256, 0, stream>>>(h5, batch, gmax, gsum, gcnt);
    final_kernel<<<(G_BATCH * OUTD) / 256, 256, 0, stream>>>(gmax, gsum, gcnt, out);
}